// SAGEConv2Layer_80547816669346
// MI455X (gfx1250) — compile-verified
//
#include <hip/hip_runtime.h>

typedef float v8f __attribute__((ext_vector_type(8)));
typedef float v2f __attribute__((ext_vector_type(2)));

#define E_EDGES 500000

// ---------------------------------------------------------------------------
// Y[N,Fo] = X[N,K] @ W[K,Fo]  (fp32 WMMA 16x16x4), K compile-time constant.
// mode 0: store    mode 1: store + bias[col]    mode 2: accumulate (+=)
// Block = 128 threads = 4 waves; each wave computes 4 consecutive 16x16 M-tiles
// of one 16-wide N slab, with the whole B panel (K x 16) preloaded in VGPRs.
// A frag (16x4): lane&15 = row, (lane>>4)*2 selects the K-pair within a step.
// C/D frag     : lane&15 = col, VGPR i holds row m0 + i + 8*(lane>>4).
// ---------------------------------------------------------------------------
template <int K>
__global__ __launch_bounds__(128)
void gemm_wmma_k(const float* __restrict__ X, const float* __restrict__ W,
                 const float* __restrict__ bias, float* __restrict__ Y,
                 int Nn, int Fo, int mode)
{
    constexpr int KS = K / 4;
    const int lane = threadIdx.x & 31;
    const int wave = threadIdx.x >> 5;
    const int half = lane >> 4;
    const int l15  = lane & 15;
    const int col  = blockIdx.y * 16 + l15;

    // Preload all B fragments for this 16-column slab (K/4 x v2f registers).
    v2f b[KS];
#pragma unroll
    for (int j = 0; j < KS; ++j) {
        const int ka = 4 * j + 2 * half;
        b[j].x = W[ka * Fo + col];
        b[j].y = W[(ka + 1) * Fo + col];
    }
    const float bv = (mode == 1) ? bias[col] : 0.0f;

#pragma unroll
    for (int tile = 0; tile < 4; ++tile) {
        const int m0 = (blockIdx.x * 16 + wave * 4 + tile) * 16;
        if (m0 >= Nn) break;
        int row = m0 + l15;
        if (row > Nn - 1) row = Nn - 1;        // clamp tail rows: no divergent loads
        const float* xr = X + (long long)row * K + 2 * half;

        v8f c = {};
#pragma unroll
        for (int j = 0; j < KS; ++j) {
            v2f a;
            a.x = xr[4 * j];
            a.y = xr[4 * j + 1];
            c = __builtin_amdgcn_wmma_f32_16x16x4_f32(false, a, false, b[j],
                                                      (short)0, c, false, false);
        }
#pragma unroll
        for (int i = 0; i < 8; ++i) {
            const int m = m0 + i + 8 * half;
            if (m < Nn) {
                float* yp = Y + (long long)m * Fo + col;
                const float v = c[i] + bv;
                if (mode == 2) *yp += v; else *yp = v;
            }
        }
    }
}

static inline void launch_gemm(const float* X, const float* W, const float* bias,
                               float* Y, int Nn, int K, int Fo, int mode,
                               hipStream_t s)
{
    dim3 g((Nn + 255) / 256, Fo / 16);
    if (K == 64)      gemm_wmma_k<64><<<g, 128, 0, s>>>(X, W, bias, Y, Nn, Fo, mode);
    else if (K == 48) gemm_wmma_k<48><<<g, 128, 0, s>>>(X, W, bias, Y, Nn, Fo, mode);
    else              gemm_wmma_k<32><<<g, 128, 0, s>>>(X, W, bias, Y, Nn, Fo, mode);
}

// dst[i] += src[i]  (tiny weight-matrix summation across relations)
__global__ void acc_add(float* __restrict__ dst, const float* __restrict__ src, int n)
{
    int i = blockIdx.x * blockDim.x + threadIdx.x;
    if (i < n) dst[i] += src[i];
}

// Per-edge row scatter: agg[dst] += z[src] (Fo floats, float4 chunks), cnt[dst] += 1
__global__ void scatter_rows(const int* __restrict__ sIdx, const int* __restrict__ dIdx,
                             const float* __restrict__ z, float* __restrict__ agg,
                             float* __restrict__ cnt, int E, int Fo)
{
    const int chunks = Fo >> 2;
    const long long t = (long long)blockIdx.x * blockDim.x + threadIdx.x;
    if (t >= (long long)E * chunks) return;
    const int e  = (int)(t / chunks);
    const int ci = (int)(t % chunks);
    const int s = sIdx[e], d = dIdx[e];
    const float4 v = ((const float4*)(z + (long long)s * Fo))[ci];
    float* ap = agg + (long long)d * Fo + ci * 4;
    atomicAdd(ap + 0, v.x); atomicAdd(ap + 1, v.y);
    atomicAdd(ap + 2, v.z); atomicAdd(ap + 3, v.w);
    if (ci == 0) atomicAdd(cnt + d, 1.0f);
}

// Scalar scatter (layer 2, Fo=1)
__global__ void scatter_scalar(const int* __restrict__ sIdx, const int* __restrict__ dIdx,
                               const float* __restrict__ z, float* __restrict__ agg,
                               float* __restrict__ cnt, int E)
{
    int e = blockIdx.x * blockDim.x + threadIdx.x;
    if (e >= E) return;
    atomicAdd(agg + dIdx[e], z[sIdx[e]]);
    atomicAdd(cnt + dIdx[e], 1.0f);
}

// h[n,f] += agg[n,f] / max(cnt[n], 1)
__global__ void accum_mean(const float* __restrict__ agg, const float* __restrict__ cnt,
                           float* __restrict__ h, int Nn, int Fo)
{
    const long long t = (long long)blockIdx.x * blockDim.x + threadIdx.x;
    if (t >= (long long)Nn * Fo) return;
    const int n = (int)(t / Fo);
    float c = cnt[n];
    c = c > 1.0f ? c : 1.0f;
    h[t] += agg[t] / c;
}

__global__ void sigmoid_ip(float* __restrict__ h, long long n)
{
    const long long t = (long long)blockIdx.x * blockDim.x + threadIdx.x;
    if (t < n) h[t] = 1.0f / (1.0f + __expf(-h[t]));
}

__global__ void sigmoid_to(const float* __restrict__ a, float* __restrict__ o, int n)
{
    int t = blockIdx.x * blockDim.x + threadIdx.x;
    if (t < n) o[t] = 1.0f / (1.0f + __expf(-a[t]));
}

// y[n] = dot(X[n,:], w[0:K]) (+ bsum[0] if bsum != null)
__global__ void gemv_f32(const float* __restrict__ X, const float* __restrict__ w,
                         const float* __restrict__ bsum, float* __restrict__ y,
                         int Nn, int K)
{
    int n = blockIdx.x * blockDim.x + threadIdx.x;
    if (n >= Nn) return;
    float acc = bsum ? bsum[0] : 0.0f;
    const float* xr = X + (long long)n * K;
    for (int k = 0; k < K; k += 4)
        acc += xr[k] * w[k] + xr[k + 1] * w[k + 1]
             + xr[k + 2] * w[k + 2] + xr[k + 3] * w[k + 3];
    y[n] = acc;
}

// ---------------------------------------------------------------------------
extern "C" void kernel_launch(void* const* d_in, const int* in_sizes, int n_in,
                              void* d_out, int out_size, void* d_ws, size_t ws_size,
                              hipStream_t stream)
{
    (void)in_sizes; (void)out_size; (void)ws_size;
    static const int NT[3] = {200000, 50000, 20000};
    static const int FT[3] = {64, 48, 32};
    const int E = E_EDGES;

    // rel tables: {ei input idx, sorted-param idx, src type, dst type}
    struct Rel { int ei, ps, s, d; };
    static const Rel R1[18] = {
        {3,9,0,0},{4,16,1,0},{5,2,2,0},{6,10,0,1},{7,17,1,1},{8,3,2,1},
        {9,8,0,2},{10,15,1,2},{11,7,0,1},{12,5,0,0},{13,13,1,0},{14,0,2,0},
        {15,6,0,1},{16,14,1,1},{17,1,2,1},{18,4,0,2},{19,12,1,2},{20,11,1,0}};
    static const int S1[18][2] = {  // sorted-order (src,dst) for params1
        {2,0},{2,1},{2,0},{2,1},{0,2},{0,0},{0,1},{0,1},{0,2},
        {0,0},{0,1},{1,0},{1,2},{1,0},{1,1},{1,2},{1,0},{1,1}};
    static const Rel R2[14] = {
        {3,7,0,0},{4,12,1,0},{5,2,2,0},{6,8,0,1},{7,13,1,1},{8,3,2,1},
        {11,6,0,1},{12,4,0,0},{13,10,1,0},{14,0,2,0},{15,5,0,1},{16,11,1,1},
        {17,1,2,1},{20,9,1,0}};
    static const int S2[14][2] = {
        {2,0},{2,1},{2,0},{2,1},{0,0},{0,1},{0,1},{0,0},{0,1},
        {1,0},{1,0},{1,1},{1,0},{1,1}};

    const float* x[3] = {(const float*)d_in[0], (const float*)d_in[1], (const float*)d_in[2]};

    const float *Wl1[18], *Wr1[18], *b1[18];
    const float *Wl2[14], *Wr2[14], *b2[14];
    if (n_in >= 118) {
        // each param leaf is its own input pointer (pytree-flattened: Wl, Wr, b per key)
        for (int i = 0; i < 18; ++i) {
            int base = 21 + 3 * R1[i].ps;
            Wl1[i] = (const float*)d_in[base];
            Wr1[i] = (const float*)d_in[base + 1];
            b1[i]  = (const float*)d_in[base + 2];
        }
        for (int i = 0; i < 14; ++i) {
            int base = 75 + 3 * R2[i].ps;
            Wl2[i] = (const float*)d_in[base];
            Wr2[i] = (const float*)d_in[base + 1];
            b2[i]  = (const float*)d_in[base + 2];
        }
    } else {
        // params1/params2 delivered as two concatenated buffers (sorted-key leaf order)
        const float* p1 = (const float*)d_in[21];
        const float* p2 = (const float*)d_in[22];
        long long off1[18][3]; long long off = 0;
        for (int s = 0; s < 18; ++s) {
            int Fs = FT[S1[s][0]], Fo = FT[S1[s][1]];
            off1[s][0] = off; off += (long long)Fs * Fo;   // Wl
            off1[s][1] = off; off += (long long)Fo * Fo;   // Wr ([Fd,Fo], Fd==Fo layer1)
            off1[s][2] = off; off += Fo;                   // b
        }
        long long off2[14][3]; off = 0;
        for (int s = 0; s < 14; ++s) {
            int Fs = FT[S2[s][0]], Fd = FT[S2[s][1]];
            off2[s][0] = off; off += Fs;  // Wl [Fs,1]
            off2[s][1] = off; off += Fd;  // Wr [Fd,1]
            off2[s][2] = off; off += 1;   // b
        }
        for (int i = 0; i < 18; ++i) {
            int s = R1[i].ps;
            Wl1[i] = p1 + off1[s][0]; Wr1[i] = p1 + off1[s][1]; b1[i] = p1 + off1[s][2];
        }
        for (int i = 0; i < 14; ++i) {
            int s = R2[i].ps;
            Wl2[i] = p2 + off2[s][0]; Wr2[i] = p2 + off2[s][1]; b2[i] = p2 + off2[s][2];
        }
    }

    // ---- workspace layout (all fp32) -------------------------------------
    float* p = (float*)d_ws;
    float* h[3];
    h[0] = p; p += (long long)200000 * 64;
    h[1] = p; p += (long long)50000 * 48;
    h[2] = p; p += (long long)20000 * 32;
    float* z    = p; p += (long long)200000 * 64;   // per-relation z = x_src @ Wl
    float* agg  = p; p += (long long)200000 * 64;   // per-relation segment sums of z
    float* cnt  = p; p += 200000;
    float* wsum = p; p += 64 * 64;                  // summed root weights per dst type
    float* bsum = p; p += 64;
    float* z2   = p; p += 200000;
    float* agg2 = p; p += 200000;
    float* cnt2 = p; p += 200000;
    float* acc2[2];
    acc2[0] = p; p += 200000;
    acc2[1] = p; p += 50000;

    // ======================= LAYER 1 ======================================
    // Root terms: h[t] = x[t] @ (sum_r Wr_r) + (sum_r b_r)  — Wr commutes with HeteroConv sum
    for (int t = 0; t < 3; ++t) {
        const int Fo = FT[t];
        hipMemsetAsync(wsum, 0, (size_t)Fo * Fo * sizeof(float), stream);
        hipMemsetAsync(bsum, 0, (size_t)Fo * sizeof(float), stream);
        for (int i = 0; i < 18; ++i) {
            if (R1[i].d != t) continue;
            acc_add<<<(Fo * Fo + 255) / 256, 256, 0, stream>>>(wsum, Wr1[i], Fo * Fo);
            acc_add<<<1, 64, 0, stream>>>(bsum, b1[i], Fo);
        }
        launch_gemm(x[t], wsum, bsum, h[t], NT[t], Fo, Fo, 1, stream);
    }

    // Neighbor terms: z = x_src @ Wl (WMMA), scatter-mean of z, h[dst] += mean
    for (int i = 0; i < 18; ++i) {
        const int st = R1[i].s, dt = R1[i].d;
        const int Ns = NT[st], Nd = NT[dt], Ks = FT[st], Fo = FT[dt];
        const int* ei = (const int*)d_in[R1[i].ei];

        hipMemsetAsync(agg, 0, (size_t)Nd * Fo * sizeof(float), stream);
        hipMemsetAsync(cnt, 0, (size_t)Nd * sizeof(float), stream);

        launch_gemm(x[st], Wl1[i], nullptr, z, Ns, Ks, Fo, 0, stream);

        long long tot = (long long)E * (Fo >> 2);
        scatter_rows<<<(unsigned)((tot + 255) / 256), 256, 0, stream>>>(
            ei, ei + E, z, agg, cnt, E, Fo);

        long long th = (long long)Nd * Fo;
        accum_mean<<<(unsigned)((th + 255) / 256), 256, 0, stream>>>(agg, cnt, h[dt], Nd, Fo);
    }

    // x1 = sigmoid(h)  (in place)
    for (int t = 0; t < 3; ++t) {
        long long n = (long long)NT[t] * FT[t];
        sigmoid_ip<<<(unsigned)((n + 255) / 256), 256, 0, stream>>>(h[t], n);
    }

    // ======================= LAYER 2 (Fo = 1) =============================
    // Root terms: acc2[t] = x1[t] . (sum_r Wr2_r) + sum_r b2_r
    for (int t = 0; t < 2; ++t) {
        const int K = FT[t];
        hipMemsetAsync(wsum, 0, (size_t)K * sizeof(float), stream);
        hipMemsetAsync(bsum, 0, sizeof(float), stream);
        for (int i = 0; i < 14; ++i) {
            if (R2[i].d != t) continue;
            acc_add<<<1, 64, 0, stream>>>(wsum, Wr2[i], K);
            acc_add<<<1, 32, 0, stream>>>(bsum, b2[i], 1);
        }
        gemv_f32<<<(NT[t] + 255) / 256, 256, 0, stream>>>(h[t], wsum, bsum, acc2[t], NT[t], K);
    }

    // Neighbor terms: z2 = x1_src . Wl2 (scalar per node), then scalar scatter-mean
    for (int i = 0; i < 14; ++i) {
        const int st = R2[i].s, dt = R2[i].d;
        const int Ns = NT[st], Nd = NT[dt], Ks = FT[st];
        const int* ei = (const int*)d_in[R2[i].ei];

        hipMemsetAsync(agg2, 0, (size_t)Nd * sizeof(float), stream);
        hipMemsetAsync(cnt2, 0, (size_t)Nd * sizeof(float), stream);

        gemv_f32<<<(Ns + 255) / 256, 256, 0, stream>>>(h[st], Wl2[i], nullptr, z2, Ns, Ks);
        scatter_scalar<<<(E + 255) / 256, 256, 0, stream>>>(ei, ei + E, z2, agg2, cnt2, E);
        accum_mean<<<(Nd + 255) / 256, 256, 0, stream>>>(agg2, cnt2, acc2[dt], Nd, 1);
    }

    // Final sigmoid -> d_out  (pred_ind ++ pred_org)
    float* out = (float*)d_out;
    sigmoid_to<<<(200000 + 255) / 256, 256, 0, stream>>>(acc2[0], out, 200000);
    sigmoid_to<<<(50000 + 255) / 256, 256, 0, stream>>>(acc2[1], out + 200000, 50000);
}